// GraphConv_4870492914285
// MI455X (gfx1250) — compile-verified
//
#include <hip/hip_runtime.h>
#include <hip/hip_bf16.h>

// GraphConv: out = segment_sum(vals * (X@W)[cols], rows) + b
// Phase 1: support = X@W. X tile staged into LDS by the Tensor Data Mover
//          (tensor_load_to_lds + s_wait_tensorcnt, hardware-padded layout),
//          compute via V_WMMA_F32_16X16X4_F32 (fp32-exact matrix path),
//          B fragments streamed from the L2-resident 128 KB W.
// Phase 2: out = b, then edge-parallel gather + atomic scatter-add
//          (support/out both fit in 192MB L2).

#define N_NODES 50000
#define N_EDGES 800000
#define IN_DIM  256
#define OUT_DIM 128

typedef __attribute__((ext_vector_type(2))) float v2f;
typedef __attribute__((ext_vector_type(8))) float v8f;
typedef unsigned int v4u __attribute__((ext_vector_type(4)));
typedef int v4i __attribute__((ext_vector_type(4)));
typedef int v8i __attribute__((ext_vector_type(8)));

// X tile row stride in floats: 260 % 64 == 4 so the 16 A-fragment lanes land
// on distinct LDS banks. TDM reproduces this via pad_interval=7 (pad after
// every 256 DWORDs) and pad_amount=3 (insert 4 DWORDs).
#define XS_STRIDE 260

// Build D# group0: count=1, lds byte addr, 57-bit global byte addr, type=2.
__device__ inline v4u tdm_group0(unsigned lds_byte_off, unsigned long long gaddr) {
  v4u g;
  g.x = 1u;                                   // count=1, no gather, user mode
  g.y = lds_byte_off;                         // lds_addr
  g.z = (unsigned)(gaddr & 0xFFFFFFFFull);    // global_addr[31:0]
  g.w = (unsigned)((gaddr >> 32) & 0x1FFFFFFu) | (2u << 30);  // addr[56:32], type=2
  return g;
}

// Build D# group1 for a 2D tile: dim0/dim1, tile0/tile1, row stride, optional
// LDS padding. data_size=2 -> 4-byte elements.
__device__ inline v8i tdm_group1_2d(unsigned dim0, unsigned dim1,
                                    unsigned tile0, unsigned tile1,
                                    unsigned stride0,
                                    unsigned pad_en, unsigned pad_interval,
                                    unsigned pad_amount) {
  v8i g;
  g[0] = (int)((2u << 16) | (pad_en << 20) | (pad_interval << 22) | (pad_amount << 25));
  g[1] = (int)((dim0 & 0xFFFFu) << 16);               // [15:0] atomic_bar=0, [31:16] dim0.lo
  g[2] = (int)(((dim0 >> 16) & 0xFFFFu) | ((dim1 & 0xFFFFu) << 16));
  g[3] = (int)(((dim1 >> 16) & 0xFFFFu) | ((tile0 & 0xFFFFu) << 16));
  g[4] = (int)(tile1 & 0xFFFFu);                      // tile_dim1, tile_dim2=0
  g[5] = (int)stride0;                                // tensor_dim0_stride[31:0]
  g[6] = 0;                                           // stride0[47:32], dim1_stride.lo
  g[7] = 0;
  return g;
}

__device__ inline void tdm_load(v4u g0, v8i g1) {
  v4i z4 = {};
#if __clang_major__ >= 23
  v8i z8 = {};
  __builtin_amdgcn_tensor_load_to_lds(g0, g1, z4, z4, z8, 0);
#else
  __builtin_amdgcn_tensor_load_to_lds(g0, g1, z4, z4, 0);
#endif
}

__global__ __launch_bounds__(256) void gemm_support_kernel(
    const float* __restrict__ x, const float* __restrict__ W,
    float* __restrict__ support) {
  __shared__ float xs[16 * XS_STRIDE];

  const size_t rowBase = (size_t)blockIdx.x * 16;

  // ptrtoint escapes xs (so the compiler cannot prove it unwritten) and its
  // low 32 bits are the wave-relative LDS byte offset the D# wants.
  const unsigned xs_lds = (unsigned)(uintptr_t)&xs[0];

  if (threadIdx.x < 32) {  // one wave drives the TDM
    // X tile: 256 wide x 16 rows, row stride 256, pad 4 DWORDs / 256 DWORDs.
    tdm_load(tdm_group0(xs_lds,
                        (unsigned long long)(uintptr_t)(x + rowBase * IN_DIM)),
             tdm_group1_2d(IN_DIM, 16, IN_DIM, 16, IN_DIM, 1u, 7u, 3u));
    __builtin_amdgcn_s_wait_tensorcnt(0);
  }
  __syncthreads();
  // TDM wrote LDS behind the compiler's back: force it to reload from LDS.
  asm volatile("" : : "s"(xs_lds) : "memory");

  const int lane = threadIdx.x & 31;
  const int wave = threadIdx.x >> 5;   // N-tile id 0..7
  const int m    = lane & 15;          // A row / B,C,D column-in-tile
  const int hi   = lane >> 4;          // lane-half selects K sub-block
  const int ncol = wave * 16 + m;      // global output column

  __builtin_prefetch(&W[ncol], 0, 3);  // global_prefetch_b8 (W slab is L2-hot)

  v8f acc = {};
#pragma unroll 8
  for (int k0 = 0; k0 < IN_DIM; k0 += 4) {
    const int k = k0 + 2 * hi;
    // A 16x4 f32 fragment: lane m holds K = {k, k+1} of row m (ISA layout).
    v2f a = *(const v2f*)&xs[m * XS_STRIDE + k];
    // B 4x16 f32 fragment: lane holds rows K={k,k+1} at column ncol.
    v2f b;
    b.x = W[(size_t)k * OUT_DIM + ncol];
    b.y = W[(size_t)(k + 1) * OUT_DIM + ncol];
    acc = __builtin_amdgcn_wmma_f32_16x16x4_f32(
        /*neg_a=*/false, a, /*neg_b=*/false, b,
        /*c_mod=*/(short)0, acc, /*reuse_a=*/false, /*reuse_b=*/false);
  }

  // D 16x16 f32 layout: VGPR i -> row i (lanes 0-15) / row 8+i (lanes 16-31).
  float* dst = support + rowBase * OUT_DIM;
#pragma unroll
  for (int i = 0; i < 8; ++i) {
    const int row = i + 8 * hi;
    dst[(size_t)row * OUT_DIM + ncol] = acc[i];
  }
}

__global__ __launch_bounds__(256) void init_out_kernel(
    const float* __restrict__ b, float* __restrict__ out, int total) {
  const int i = blockIdx.x * 256 + threadIdx.x;
  if (i < total) out[i] = b[i & (OUT_DIM - 1)];
}

__global__ __launch_bounds__(256) void edge_scatter_kernel(
    const int* __restrict__ rows, const int* __restrict__ cols,
    const float* __restrict__ vals, const float* __restrict__ support,
    float* __restrict__ out) {
  const int e = blockIdx.x * 8 + (threadIdx.x >> 5);  // one wave32 per edge
  if (e >= N_EDGES) return;
  const int lane = threadIdx.x & 31;
  const int r = rows[e];
  const int c = cols[e];
  const float v = vals[e];

  // 512 B coalesced gather of support[c][:] (L2-resident), scaled scatter-add.
  const float4 m4 = ((const float4*)(support + (size_t)c * OUT_DIM))[lane];
  float* dst = out + (size_t)r * OUT_DIM + lane * 4;
  atomicAdd(dst + 0, v * m4.x);
  atomicAdd(dst + 1, v * m4.y);
  atomicAdd(dst + 2, v * m4.z);
  atomicAdd(dst + 3, v * m4.w);
}

extern "C" void kernel_launch(void* const* d_in, const int* in_sizes, int n_in,
                              void* d_out, int out_size, void* d_ws, size_t ws_size,
                              hipStream_t stream) {
  const int*   adj_rows = (const int*)d_in[0];
  const int*   adj_cols = (const int*)d_in[1];
  const float* adj_vals = (const float*)d_in[2];
  const float* x        = (const float*)d_in[3];
  const float* W        = (const float*)d_in[4];
  const float* b        = (const float*)d_in[5];
  float* out     = (float*)d_out;
  float* support = (float*)d_ws;  // needs 50000*128*4 = 25.6 MB scratch

  // support = X @ W   (3125 blocks x 8 waves, one 16x16 WMMA tile per wave)
  gemm_support_kernel<<<N_NODES / 16, 256, 0, stream>>>(x, W, support);

  // out = broadcast(b)
  init_out_kernel<<<(N_NODES * OUT_DIM + 255) / 256, 256, 0, stream>>>(
      b, out, N_NODES * OUT_DIM);

  // out += segment_sum(vals * support[cols])
  edge_scatter_kernel<<<N_EDGES / 8, 256, 0, stream>>>(
      adj_rows, adj_cols, adj_vals, support, out);
}